// encoder_52329881534785
// MI455X (gfx1250) — compile-verified
//
#include <hip/hip_runtime.h>
#include <hip/hip_bf16.h>
#include <math.h>

// ---------------------------------------------------------------------------
// LSTM over T=2048, H=1024 on gfx1250 (MI455X).
//  Phase 1: gx[T,4H] = emb[tokens] @ W_ih^T + (b_ih+b_hh)  -> fp32 WMMA GEMM
//  Phase 2: sequential scan, persistent kernel, 1 grid barrier / step,
//           W_hh streamed from L2 (16 MB, resident), c held in LDS.
// ---------------------------------------------------------------------------

typedef __attribute__((ext_vector_type(2))) float v2f;
typedef __attribute__((ext_vector_type(8))) float v8f;

#define TSEQ 2048
#define HDIM 1024
#define G4H  4096
#define NWG  128              // persistent scan workgroups
#define HPW  (HDIM / NWG)     // 8 h elements per workgroup
#define ROWS (4 * HPW)        // 32 gate rows per workgroup

// ---------------- init: zero h double-buffer + barrier state ----------------
__global__ void lstm_init_kernel(float* hbuf, unsigned* sync) {
  int tid = blockIdx.x * blockDim.x + threadIdx.x;
  if (tid < 2 * HDIM) hbuf[tid] = 0.0f;
  if (tid < 2) sync[tid] = 0u;
}

// ---------------- Phase 1: gx GEMM via V_WMMA_F32_16X16X4_F32 ----------------
// One wave computes a 16(M=t) x 64(N) strip: A fragment reused across 4 N-tiles.
__global__ __launch_bounds__(256)
void gx_gemm_kernel(const int* __restrict__ tokens,
                    const float* __restrict__ emb,
                    const float* __restrict__ Wih,
                    const float* __restrict__ bih,
                    const float* __restrict__ bhh,
                    float* __restrict__ gx) {
  const int wave = (blockIdx.x * blockDim.x + threadIdx.x) >> 5;
  const int lane = threadIdx.x & 31;
  const int tTile = wave >> 6;   // 128 tiles of 16 timesteps
  const int nGrp  = wave & 63;   // 64 groups of 64 output columns
  const int m  = lane & 15;      // M (and N) index within tile
  const int kh = lane >> 4;      // selects K pair {0,1} vs {2,3}

  const int t0 = tTile * 16;
  const int n0 = nGrp * 64;

  // A: row of gathered embedding (M = t0+m), contiguous K pair
  const float* arow = emb + (size_t)tokens[t0 + m] * HDIM + 2 * kh;
  // B[k][n] = W_ih[n][k]: contiguous K pair from W_ih row n
  const float* brow[4];
#pragma unroll
  for (int j = 0; j < 4; ++j)
    brow[j] = Wih + (size_t)(n0 + 16 * j + m) * HDIM + 2 * kh;

  v8f acc[4];
#pragma unroll
  for (int j = 0; j < 4; ++j) acc[j] = (v8f){0.f,0.f,0.f,0.f,0.f,0.f,0.f,0.f};

#pragma unroll 2
  for (int k = 0; k < HDIM; k += 4) {
    v2f a = *(const v2f*)(arow + k);
#pragma unroll
    for (int j = 0; j < 4; ++j) {
      v2f b = *(const v2f*)(brow[j] + k);
      acc[j] = __builtin_amdgcn_wmma_f32_16x16x4_f32(
          /*neg_a=*/false, a, /*neg_b=*/false, b,
          /*c_mod=*/(short)0, acc[j], /*reuse_a=*/false, /*reuse_b=*/false);
    }
  }

  // epilogue: bias add + store.  D layout: VGPR v -> M = v + 8*kh, N = lane&15
#pragma unroll
  for (int j = 0; j < 4; ++j) {
    const int n = n0 + 16 * j + m;
    const float bias = bih[n] + bhh[n];
#pragma unroll
    for (int v = 0; v < 8; ++v) {
      const int r = t0 + v + 8 * kh;
      gx[(size_t)r * G4H + n] = acc[j][v] + bias;
    }
  }
}

// ---------------- Phase 2: persistent recurrent scan ----------------
__global__ __launch_bounds__(256)
void lstm_scan_kernel(const float* __restrict__ Whh,
                      const float* __restrict__ gx,
                      float* __restrict__ hbuf,    // [2][HDIM] double buffer
                      unsigned* __restrict__ sync, // {count, generation}
                      float* __restrict__ out) {   // [2*HDIM] = h ++ c
  __shared__ __align__(16) float sh[HDIM];   // staged h_{t-1}
  __shared__ float part[256];                // dot partials
  __shared__ float gate[ROWS];               // 32 gate pre-activations
  __shared__ float cst[HPW];                 // resident cell state slice

  const int tid = threadIdx.x;
  const int w   = blockIdx.x;
  const int row = tid & 31;                  // gate row within block
  const int seg = tid >> 5;                  // 8 K-segments of 128
  // global gate row: gate = row>>3 in {i,f,g,o}, element = w*HPW + (row&7)
  const int grow = (row >> 3) * HDIM + w * HPW + (row & 7);
  const float* wrow = Whh + (size_t)grow * HDIM + seg * 128;  // L2-resident
  const float* gxp  = gx + grow;

  if (tid < HPW) cst[tid] = 0.0f;

  for (int t = 0; t < TSEQ; ++t) {
    const float* hin  = hbuf + (size_t)(t & 1) * HDIM;
    float*       hout = hbuf + (size_t)((t + 1) & 1) * HDIM;

    // stage h_{t-1} into LDS (256 threads x float4)
    *(float4*)&sh[tid * 4] = *(const float4*)&hin[tid * 4];
    __syncthreads();

    // partial dot: 32 rows x 8 segments, 128 MACs each
    float s = 0.f;
#pragma unroll 8
    for (int i = 0; i < 32; ++i) {
      float4 w4 = *(const float4*)(wrow + i * 4);
      float4 h4 = *(const float4*)&sh[seg * 128 + i * 4];
      s = fmaf(w4.x, h4.x, s); s = fmaf(w4.y, h4.y, s);
      s = fmaf(w4.z, h4.z, s); s = fmaf(w4.w, h4.w, s);
    }
    part[tid] = s;
    __syncthreads();

    if (tid < ROWS) {
      float g = gxp[(size_t)t * G4H];
#pragma unroll
      for (int q = 0; q < 8; ++q) g += part[tid + 32 * q];
      gate[tid] = g;
    }
    __syncthreads();

    if (tid < HPW) {
      const float gi = gate[tid];
      const float gf = gate[HPW + tid];
      const float gg = gate[2 * HPW + tid];
      const float go = gate[3 * HPW + tid];
      const float is = 1.f / (1.f + __expf(-gi));
      const float fs = 1.f / (1.f + __expf(-gf));
      const float gt = tanhf(gg);
      const float os = 1.f / (1.f + __expf(-go));
      const float c  = fs * cst[tid] + is * gt;
      cst[tid] = c;
      hout[w * HPW + tid] = os * tanhf(c);
    }
    __syncthreads();

    // grid barrier (release/acquire through L2, sense via generation counter)
    if (tid == 0) {
      __threadfence();
      const unsigned g =
          __hip_atomic_load(&sync[1], __ATOMIC_RELAXED, __HIP_MEMORY_SCOPE_AGENT);
      if (atomicAdd(&sync[0], 1u) == (unsigned)(NWG - 1)) {
        __hip_atomic_store(&sync[0], 0u, __ATOMIC_RELAXED, __HIP_MEMORY_SCOPE_AGENT);
        __hip_atomic_store(&sync[1], g + 1u, __ATOMIC_RELEASE, __HIP_MEMORY_SCOPE_AGENT);
      } else {
        while (__hip_atomic_load(&sync[1], __ATOMIC_ACQUIRE,
                                 __HIP_MEMORY_SCOPE_AGENT) == g)
          __builtin_amdgcn_s_sleep(1);
      }
      __threadfence();
    }
    __syncthreads();
  }

  // final h landed in parity (TSEQ & 1) == 0; this block wrote its own slice
  if (tid < HPW) {
    out[w * HPW + tid]        = hbuf[w * HPW + tid];
    out[HDIM + w * HPW + tid] = cst[tid];
  }
}

// ---------------------------------------------------------------------------
extern "C" void kernel_launch(void* const* d_in, const int* in_sizes, int n_in,
                              void* d_out, int out_size, void* d_ws, size_t ws_size,
                              hipStream_t stream) {
  const int*   tokens = (const int*)d_in[0];
  const float* emb    = (const float*)d_in[1];
  const float* Wih    = (const float*)d_in[2];
  const float* Whh    = (const float*)d_in[3];
  const float* bih    = (const float*)d_in[4];
  const float* bhh    = (const float*)d_in[5];
  float* out = (float*)d_out;

  char* ws = (char*)d_ws;
  const size_t gx_bytes = (size_t)TSEQ * G4H * sizeof(float);   // 32 MB
  float*    gx   = (float*)ws;
  float*    hbuf = (float*)(ws + gx_bytes);                     // 2*HDIM floats
  unsigned* sync = (unsigned*)(ws + gx_bytes + 2 * HDIM * sizeof(float));

  lstm_init_kernel<<<dim3((2 * HDIM + 255) / 256), 256, 0, stream>>>(hbuf, sync);
  // 128 t-tiles * 64 n-groups = 8192 wave tasks -> 1024 blocks x 8 waves
  gx_gemm_kernel<<<dim3(1024), 256, 0, stream>>>(tokens, emb, Wih, bih, bhh, gx);
  lstm_scan_kernel<<<dim3(NWG), 256, 0, stream>>>(Whh, gx, hbuf, sync, out);
}